// BGRL_28544352649385
// MI455X (gfx1250) — compile-verified
//
#include <hip/hip_runtime.h>

#define DD 128
#define NROWS 50000
#define NEDGE 800000
#define ND (NROWS * DD)
#define BN_EPS 1e-5f

typedef __attribute__((ext_vector_type(2))) float v2f;
typedef __attribute__((ext_vector_type(8))) float v8f;

// ---------------- utility kernels ----------------
__global__ void k_zero_f32(float* p, int n) {
    int i = blockIdx.x * 256 + threadIdx.x;
    if (i < n) p[i] = 0.0f;
}
__global__ void k_zero_i32(int* p, int n) {
    int i = blockIdx.x * 256 + threadIdx.x;
    if (i < n) p[i] = 0;
}
__global__ void k_ewise_add4(const float* __restrict__ a, const float* __restrict__ b,
                             float* __restrict__ o, int n4) {
    int i = blockIdx.x * 256 + threadIdx.x;
    if (i >= n4) return;
    const float4* av = reinterpret_cast<const float4*>(a);
    const float4* bv = reinterpret_cast<const float4*>(b);
    float4* ov = reinterpret_cast<float4*>(o);
    float4 x = av[i], y = bv[i];
    float4 r;
    r.x = x.x + y.x; r.y = x.y + y.y; r.z = x.z + y.z; r.w = x.w + y.w;
    ov[i] = r;
}

// ---------------- CSR build ----------------
__global__ void k_hist(const int* __restrict__ row, int* __restrict__ counts, int e) {
    int i = blockIdx.x * 256 + threadIdx.x;
    if (i < e) atomicAdd(&counts[row[i]], 1);
}

__global__ __launch_bounds__(1024)
void k_scan(const int* __restrict__ counts, int* __restrict__ row_ptr,
            int* __restrict__ cursor, int n) {
    __shared__ int sh[1024];
    const int t = threadIdx.x;
    const int CH = (n + 1023) / 1024;
    const int beg = t * CH;
    const int end = (beg + CH < n) ? (beg + CH) : n;
    int s = 0;
    for (int i = beg; i < end; ++i) s += counts[i];
    sh[t] = s;
    __syncthreads();
    for (int off = 1; off < 1024; off <<= 1) {
        int v = (t >= off) ? sh[t - off] : 0;
        __syncthreads();
        sh[t] += v;
        __syncthreads();
    }
    int run = sh[t] - s; // exclusive prefix for this thread's chunk
    for (int i = beg; i < end; ++i) {
        int c = counts[i];
        row_ptr[i] = run;
        cursor[i] = run;
        run += c;
    }
    if (t == 1023) row_ptr[n] = sh[1023];
}

__global__ void k_scatter(const int* __restrict__ row, const int* __restrict__ col,
                          const float* __restrict__ val, int* __restrict__ cursor,
                          int* __restrict__ csr_col, float* __restrict__ csr_val, int e) {
    int i = blockIdx.x * 256 + threadIdx.x;
    if (i >= e) return;
    int r = row[i];
    int pos = atomicAdd(&cursor[r], 1);
    csr_col[pos] = col[i];
    csr_val[pos] = val[i];
}

// ---------------- fp32 WMMA GEMM: Out[M,128] = X[M,128] @ W[128,128] (+bias) ----
// LDS holds W in K-pair-interleaved layout: element (k,n) at
//   sW[(k>>1)*256 + n*2 + (k&1)]
// so a B fragment {W[k][n], W[k+1][n]} is one contiguous 8-byte ds_load_b64.
__global__ __launch_bounds__(128)
void k_gemm128(const float* __restrict__ X, const float* __restrict__ Wm,
               const float* __restrict__ bias, float* __restrict__ Out, int nrows) {
    __shared__ float sW[DD * DD]; // 64 KB
    const int tid = threadIdx.x;
    {
        // 64 row-pairs x 32 col-groups(float4) = 2048 tasks, 16 per thread
        #pragma unroll
        for (int i = 0; i < 16; ++i) {
            const int p = tid + i * 128;
            const int k2 = p >> 5;    // row pair 0..63
            const int g = p & 31;     // col group of 4
            const float4 r0 = *reinterpret_cast<const float4*>(Wm + (size_t)(2 * k2) * DD + g * 4);
            const float4 r1 = *reinterpret_cast<const float4*>(Wm + (size_t)(2 * k2 + 1) * DD + g * 4);
            float4 s0, s1;
            s0.x = r0.x; s0.y = r1.x; s0.z = r0.y; s0.w = r1.y;
            s1.x = r0.z; s1.y = r1.z; s1.z = r0.w; s1.w = r1.w;
            float4* d = reinterpret_cast<float4*>(&sW[k2 * 256 + g * 8]);
            d[0] = s0;
            d[1] = s1;
        }
    }
    __syncthreads();

    const int wave = tid >> 5;
    const int lane = tid & 31;
    const int tile = blockIdx.x * 4 + wave;
    const int m0 = tile * 16;
    if (m0 >= nrows) return; // wave-uniform: EXEC stays all-ones for WMMA

    const int half = lane >> 4; // 0: K pair {k0,k0+1}; 1: K pair {k0+2,k0+3}
    const int l15 = lane & 15;
    const float* xrow = X + (size_t)(m0 + l15) * DD;

    v8f acc[8];
    #pragma unroll
    for (int j = 0; j < 8; ++j) {
        v8f z = {0.f, 0.f, 0.f, 0.f, 0.f, 0.f, 0.f, 0.f};
        acc[j] = z;
    }

    for (int k0 = 0; k0 < DD; k0 += 4) {
        const int ka = k0 + 2 * half;
        // A 16x4 f32 fragment: VGPR0 = K=ka, VGPR1 = K=ka+1 (lane = row M)
        const v2f a = *reinterpret_cast<const v2f*>(xrow + ka);
        const float* sWp = sW + (ka >> 1) * 256 + l15 * 2;
        #pragma unroll
        for (int j = 0; j < 8; ++j) {
            const v2f bf = *reinterpret_cast<const v2f*>(sWp + j * 32);
            acc[j] = __builtin_amdgcn_wmma_f32_16x16x4_f32(
                false, a, false, bf, (short)0, acc[j], false, false);
        }
    }

    // C/D layout: VGPR i -> M = i (lanes 0-15) or i+8 (lanes 16-31), N = l15
    #pragma unroll
    for (int j = 0; j < 8; ++j) {
        const float bv = bias ? bias[j * 16 + l15] : 0.0f;
        #pragma unroll
        for (int i = 0; i < 8; ++i) {
            Out[(size_t)(m0 + half * 8 + i) * DD + j * 16 + l15] = acc[j][i] + bv;
        }
    }
}

// ---------------- CSR SpMM aggregation: out[r] = sum val*H[col] + bias (+x2) ----
__global__ __launch_bounds__(256)
void k_spmm(const float* __restrict__ H, const int* __restrict__ row_ptr,
            const int* __restrict__ csr_col, const float* __restrict__ csr_val,
            const float* __restrict__ bias, const float* __restrict__ addend,
            float* __restrict__ out, float* __restrict__ out2, int nrows) {
    const int r = blockIdx.x * 8 + (threadIdx.x >> 5);
    if (r >= nrows) return;
    const int lane = threadIdx.x & 31;
    const int beg = row_ptr[r];
    const int end = row_ptr[r + 1];
    float4 acc; acc.x = 0.f; acc.y = 0.f; acc.z = 0.f; acc.w = 0.f;
    int j = beg;
    // unroll by 2 to keep two 512B gathers in flight per wave
    for (; j + 1 < end; j += 2) {
        const int c0 = csr_col[j];
        const int c1 = csr_col[j + 1];
        const float v0 = csr_val[j];
        const float v1 = csr_val[j + 1];
        const float4 h0 = reinterpret_cast<const float4*>(H + (size_t)c0 * DD)[lane];
        const float4 h1 = reinterpret_cast<const float4*>(H + (size_t)c1 * DD)[lane];
        acc.x += v0 * h0.x + v1 * h1.x;
        acc.y += v0 * h0.y + v1 * h1.y;
        acc.z += v0 * h0.z + v1 * h1.z;
        acc.w += v0 * h0.w + v1 * h1.w;
    }
    if (j < end) {
        const int c = csr_col[j];
        const float v = csr_val[j];
        const float4 h = reinterpret_cast<const float4*>(H + (size_t)c * DD)[lane];
        acc.x += v * h.x; acc.y += v * h.y; acc.z += v * h.z; acc.w += v * h.w;
    }
    const float4 bv = reinterpret_cast<const float4*>(bias)[lane];
    acc.x += bv.x; acc.y += bv.y; acc.z += bv.z; acc.w += bv.w;
    reinterpret_cast<float4*>(out + (size_t)r * DD)[lane] = acc;
    if (out2) {
        const float4 av = reinterpret_cast<const float4*>(addend + (size_t)r * DD)[lane];
        float4 o2;
        o2.x = acc.x + av.x; o2.y = acc.y + av.y; o2.z = acc.z + av.z; o2.w = acc.w + av.w;
        reinterpret_cast<float4*>(out2 + (size_t)r * DD)[lane] = o2;
    }
}

// ---------------- BatchNorm column stats ----------------
__global__ __launch_bounds__(128)
void k_colstats(const float* __restrict__ U, float* __restrict__ sums,
                float* __restrict__ sumsq, int nrows) {
    const int col = threadIdx.x;
    const int r0 = blockIdx.x * 256;
    int rend = r0 + 256; if (rend > nrows) rend = nrows;
    float s = 0.f, s2 = 0.f;
    for (int r = r0; r < rend; ++r) {
        float v = U[(size_t)r * DD + col];
        s += v; s2 += v * v;
    }
    atomicAdd(&sums[col], s);
    atomicAdd(&sumsq[col], s2);
}

// ---------------- BN normalize + PReLU (in place) ----------------
__global__ void k_bn_prelu(float* __restrict__ U, const float* __restrict__ sums,
                           const float* __restrict__ sumsq, const float* __restrict__ gamma,
                           const float* __restrict__ beta, const float* __restrict__ a_ptr,
                           int nrows, int n) {
    int i = blockIdx.x * 256 + threadIdx.x;
    if (i >= n) return;
    const int col = i & (DD - 1);
    const float invN = 1.0f / (float)nrows;
    const float mu = sums[col] * invN;
    const float var = sumsq[col] * invN - mu * mu;
    const float hn = (U[i] - mu) * rsqrtf(var + BN_EPS) * gamma[col] + beta[col];
    const float a = a_ptr[0];
    U[i] = hn >= 0.f ? hn : a * hn;
}

// ---------------- BYOL row loss: acc += sum_r 2 - 2*cos(p_r, t_r) -------------
__global__ __launch_bounds__(256)
void k_row_loss(const float* __restrict__ P, const float* __restrict__ T,
                float* __restrict__ acc, int nrows) {
    __shared__ float sh[8];
    const int wave = threadIdx.x >> 5;
    const int lane = threadIdx.x & 31;
    const int r = blockIdx.x * 8 + wave;
    float pp = 0.f, tt = 0.f, pt = 0.f;
    if (r < nrows) {
        const float4 p = reinterpret_cast<const float4*>(P + (size_t)r * DD)[lane];
        const float4 t = reinterpret_cast<const float4*>(T + (size_t)r * DD)[lane];
        pp = p.x * p.x + p.y * p.y + p.z * p.z + p.w * p.w;
        tt = t.x * t.x + t.y * t.y + t.z * t.z + t.w * t.w;
        pt = p.x * t.x + p.y * t.y + p.z * t.z + p.w * t.w;
    }
    #pragma unroll
    for (int off = 16; off > 0; off >>= 1) {
        pp += __shfl_xor(pp, off, 32);
        tt += __shfl_xor(tt, off, 32);
        pt += __shfl_xor(pt, off, 32);
    }
    if (lane == 0) sh[wave] = (r < nrows) ? (2.0f - 2.0f * pt * rsqrtf(pp * tt)) : 0.0f;
    __syncthreads();
    if (threadIdx.x == 0) {
        float s = 0.f;
        #pragma unroll
        for (int i = 0; i < 8; ++i) s += sh[i];
        atomicAdd(acc, s);
    }
}

__global__ void k_finalize(float* loss, int nrows) {
    loss[0] *= (1.0f / (float)nrows);
}

// ---------------- driver ----------------
extern "C" void kernel_launch(void* const* d_in, const int* in_sizes, int n_in,
                              void* d_out, int out_size, void* d_ws, size_t ws_size,
                              hipStream_t stream) {
    const float* x      = (const float*)d_in[0];
    const float* perb   = (const float*)d_in[1];
    const int*   erow   = (const int*)d_in[2];
    const int*   ecol   = (const int*)d_in[3];
    const float* evalv  = (const float*)d_in[4];
    const float* W      = (const float*)d_in[5];
    const float* b      = (const float*)d_in[6];
    const float* Wt     = (const float*)d_in[7];
    const float* bt     = (const float*)d_in[8];
    const float* W1     = (const float*)d_in[9];
    const float* b1     = (const float*)d_in[10];
    const float* gamma  = (const float*)d_in[11];
    const float* beta   = (const float*)d_in[12];
    const float* prelua = (const float*)d_in[13];
    const float* W2     = (const float*)d_in[14];
    const float* b2     = (const float*)d_in[15];

    float* embed = (float*)d_out;       // [N*D]
    float* loss  = embed + ND;          // scalar

    // workspace layout
    float* X2  = (float*)d_ws;          // N*D
    float* H   = X2 + ND;               // N*D (GEMM temp, reused)
    float* ONX = H + ND;                // N*D online_x -> later P1
    float* ONY = ONX + ND;              // N*D online_y -> later P2
    float* TGX = ONY + ND;              // N*D target_x
    float* TGY = TGX + ND;              // N*D target_y
    int* row_ptr = (int*)(TGY + ND);    // N+1
    int* counts  = row_ptr + (NROWS + 1);
    int* cursor  = counts + NROWS;
    int* csr_col = cursor + NROWS;      // E
    float* csr_val = (float*)(csr_col + NEDGE); // E
    float* sums  = csr_val + NEDGE;     // D
    float* sumsq = sums + DD;           // D

    const int gemmGrid = (NROWS / 16 + 3) / 4;   // 3125 tiles, 4 waves/block
    const int rowGrid  = (NROWS + 7) / 8;        // wave-per-row kernels
    const int edgeGrid = (NEDGE + 255) / 256;

    // x2 = x + perb
    k_ewise_add4<<<(ND / 4 + 255) / 256, 256, 0, stream>>>(x, perb, X2, ND / 4);

    // CSR build (shared by all 4 GCN aggregations)
    k_zero_i32<<<(NROWS + 255) / 256, 256, 0, stream>>>(counts, NROWS);
    k_hist<<<edgeGrid, 256, 0, stream>>>(erow, counts, NEDGE);
    k_scan<<<1, 1024, 0, stream>>>(counts, row_ptr, cursor, NROWS);
    k_scatter<<<edgeGrid, 256, 0, stream>>>(erow, ecol, evalv, cursor, csr_col, csr_val, NEDGE);

    // zero loss accumulator (d_out is poisoned by harness)
    k_zero_f32<<<1, 256, 0, stream>>>(loss, 1);

    // GCN 1: online_x = agg(x @ W) + b
    k_gemm128<<<gemmGrid, 128, 0, stream>>>(x, W, nullptr, H, NROWS);
    k_spmm<<<rowGrid, 256, 0, stream>>>(H, row_ptr, csr_col, csr_val, b, nullptr, ONX, nullptr, NROWS);
    // GCN 2: online_y = enc_x2 = agg(x2 @ W) + b ; embed = x2 + enc_x2 (fused)
    k_gemm128<<<gemmGrid, 128, 0, stream>>>(X2, W, nullptr, H, NROWS);
    k_spmm<<<rowGrid, 256, 0, stream>>>(H, row_ptr, csr_col, csr_val, b, X2, ONY, embed, NROWS);
    // GCN 3: target_y = agg(x @ Wt) + bt
    k_gemm128<<<gemmGrid, 128, 0, stream>>>(x, Wt, nullptr, H, NROWS);
    k_spmm<<<rowGrid, 256, 0, stream>>>(H, row_ptr, csr_col, csr_val, bt, nullptr, TGY, nullptr, NROWS);
    // GCN 4: target_x = agg(x2 @ Wt) + bt
    k_gemm128<<<gemmGrid, 128, 0, stream>>>(X2, Wt, nullptr, H, NROWS);
    k_spmm<<<rowGrid, 256, 0, stream>>>(H, row_ptr, csr_col, csr_val, bt, nullptr, TGX, nullptr, NROWS);

    // predictor pass 1: px = pred(online_x); loss += byol(px, target_x)
    k_gemm128<<<gemmGrid, 128, 0, stream>>>(ONX, W1, b1, H, NROWS);
    k_zero_f32<<<1, 256, 0, stream>>>(sums, 2 * DD);
    k_colstats<<<(NROWS + 255) / 256, 128, 0, stream>>>(H, sums, sumsq, NROWS);
    k_bn_prelu<<<(ND + 255) / 256, 256, 0, stream>>>(H, sums, sumsq, gamma, beta, prelua, NROWS, ND);
    k_gemm128<<<gemmGrid, 128, 0, stream>>>(H, W2, b2, ONX, NROWS); // P1 overwrites ONX
    k_row_loss<<<rowGrid, 256, 0, stream>>>(ONX, TGX, loss, NROWS);

    // predictor pass 2: py = pred(online_y); loss += byol(py, target_y)
    k_gemm128<<<gemmGrid, 128, 0, stream>>>(ONY, W1, b1, H, NROWS);
    k_zero_f32<<<1, 256, 0, stream>>>(sums, 2 * DD);
    k_colstats<<<(NROWS + 255) / 256, 128, 0, stream>>>(H, sums, sumsq, NROWS);
    k_bn_prelu<<<(ND + 255) / 256, 256, 0, stream>>>(H, sums, sumsq, gamma, beta, prelua, NROWS, ND);
    k_gemm128<<<gemmGrid, 128, 0, stream>>>(H, W2, b2, ONY, NROWS); // P2 overwrites ONY
    k_row_loss<<<rowGrid, 256, 0, stream>>>(ONY, TGY, loss, NROWS);

    k_finalize<<<1, 1, 0, stream>>>(loss, NROWS);
}